// DigitCapsuleLayer_4475355922493
// MI455X (gfx1250) — compile-verified
//
#include <hip/hip_runtime.h>

// Capsule dynamic routing, fully fused (u_hat never materialized).
// B=1024, NIN=1152, DIN=8, NOUT=10, DOUT=16, 3 routing iterations.
// Per workgroup: 16 batches (WMMA M), 20 waves = 10 capsules x 2 i-splits.
// GEMM K=8 done as two chained V_WMMA_F32_16X16X4_F32 (exact fp32).
// All working data (x 38MB, W 6MB, b-logits 47MB) is L2-resident on MI455X.

#define B_TOT    1024
#define NIN      1152
#define DIN      8
#define NOUT     10
#define DOUT     16
#define MT       16
#define ROUTINGS 3
#define WSPLIT   2
#define WAVES    (NOUT * WSPLIT)   // 20
#define NTHREADS (WAVES * 32)      // 640
#define IH       (NIN / WSPLIT)    // 576
#define CAPS_EPS 1e-7f
#define PFDIST   16                // W prefetch distance (capsules ahead)

typedef float v8f __attribute__((ext_vector_type(8)));
typedef float v2f __attribute__((ext_vector_type(2)));

__device__ __forceinline__ float half16_sum(float t) {
    // reduction across the 16 lanes of a half-wave (xor masks stay in-half)
    t += __shfl_xor(t, 1);
    t += __shfl_xor(t, 2);
    t += __shfl_xor(t, 4);
    t += __shfl_xor(t, 8);
    return t;
}

// Multi-value butterfly: reduce t[0..7] over the 16 lanes of a half-wave with
// only 8 shuffles. On return every lane holds the full 16-lane sum of value
// index ridx(lane) = 4*bit0 + 2*bit1 + bit2 of (lane&15).
__device__ __forceinline__ float butterfly8x16(const float* t, int lane) {
    float a[4];
#pragma unroll
    for (int r = 0; r < 4; ++r) {
        float send = (lane & 1) ? t[r] : t[r + 4];
        float recv = __shfl_xor(send, 1);
        float keep = (lane & 1) ? t[r + 4] : t[r];
        a[r] = keep + recv;          // pair-sum of value (bit0*4 + r)
    }
    float b[2];
#pragma unroll
    for (int r = 0; r < 2; ++r) {
        float send = (lane & 2) ? a[r] : a[r + 2];
        float recv = __shfl_xor(send, 2);
        float keep = (lane & 2) ? a[r + 2] : a[r];
        b[r] = keep + recv;          // quad-sum of value (bit0*4 + bit1*2 + r)
    }
    {
        float send = (lane & 4) ? b[0] : b[1];
        float recv = __shfl_xor(send, 4);
        float keep = (lane & 4) ? b[1] : b[0];
        b[0] = keep + recv;          // 8-lane sum of value (bit0*4+bit1*2+bit2)
    }
    b[0] += __shfl_xor(b[0], 8);     // full 16-lane sum
    return b[0];
}

__global__ __launch_bounds__(NTHREADS) void caps_route(
    const float* __restrict__ x,      // [B, NIN, DIN]
    const float* __restrict__ W,      // [NIN, NOUT, DOUT, DIN]
    float* __restrict__ out,          // [B, NOUT, DOUT]
    float* __restrict__ blog_all)     // scratch: [B/MT][NIN][NOUT][MT]
{
    __shared__ float xch[WAVES][8][32];   // 20 KB exchange buffer

    const int tid  = threadIdx.x;
    const int wave = tid >> 5;
    const int lane = tid & 31;
    const int half = lane >> 4;     // K-half / M-half selector
    const int lm   = lane & 15;     // A-row m, B-col d, C/D col d
    const int n    = wave >> 1;     // output capsule owned by this wave
    const int sub  = wave & 1;      // which half of the i-range
    const int b0   = blockIdx.x * MT;
    const int i0   = sub * IH;
    const int i1   = i0 + IH;

    float* blog = blog_all + (size_t)blockIdx.x * ((size_t)NIN * NOUT * MT);
    // per-lane logit pointer: bp[i * NOUT*MT] = blog[m = lm, i, n]
    float* bp = blog + n * MT + lm;

    // A operand: lane holds x[b0+lm, i, e0 .. e0+1], e0 = half*2 (+4 for K-step 1)
    const float* xp = x + (size_t)(b0 + lm) * NIN * DIN + half * 2;
    // B operand: lane holds W[i, n, lm, e0 .. e0+1]
    const float* wp = W + ((size_t)n * DOUT + lm) * DIN + half * 2;

    float vcap[8];   // v[m = r + half*8][d = lm], C/D layout

    // Permuted row index produced by the butterfly reduction (agreement pass)
    const int ridx = ((lm & 1) << 2) | (lm & 2) | ((lm >> 2) & 1);

#pragma unroll
    for (int it = 0; it < ROUTINGS; ++it) {
        // ---------- softmax stats over i (axis=-2) ----------
        float mx = 0.0f, rinv = 1.0f / (float)NIN;
        if (it > 0) {
            float pm = -3.4e38f;
            for (int i = i0; i < i1; ++i)
                pm = fmaxf(pm, bp[i * (NOUT * MT)]);
            xch[wave][0][lane] = pm;
            __syncthreads();
            mx = fmaxf(pm, xch[wave ^ 1][0][lane]);
            __syncthreads();
            float ps = 0.0f;
            for (int i = i0; i < i1; ++i)
                ps += __expf(bp[i * (NOUT * MT)] - mx);
            xch[wave][0][lane] = ps;
            __syncthreads();
            ps += xch[wave ^ 1][0][lane];
            rinv = 1.0f / ps;
            __syncthreads();
        }

        // ---------- s-pass: s = sum_i diag(c[:,i]) * X_i * W_i^T ----------
        // c folded into A; the WMMA C operand chains the i-accumulation.
        v8f acc = {};
        for (int i = i0; i < i1; ++i) {
            const float* ax = xp + i * DIN;
            v2f a0 = *(const v2f*)(ax);
            v2f a1 = *(const v2f*)(ax + 4);
            float c;
            if (it == 0) c = 1.0f / (float)NIN;                       // b == 0
            else         c = __expf(bp[i * (NOUT * MT)] - mx) * rinv;
            a0 *= c;
            a1 *= c;
            const float* aw = wp + (size_t)i * (NOUT * DOUT * DIN);
            if (it == 0)   // cold pass: pull the W stream into L2 early
                __builtin_prefetch(aw + (size_t)PFDIST * (NOUT * DOUT * DIN), 0, 1);
            v2f w0 = *(const v2f*)(aw);
            v2f w1 = *(const v2f*)(aw + 4);
            acc = __builtin_amdgcn_wmma_f32_16x16x4_f32(
                      false, a0, false, w0, (short)0, acc, false, false);
            acc = __builtin_amdgcn_wmma_f32_16x16x4_f32(
                      false, a1, false, w1, (short)0, acc, false, false);
        }

        // combine the two i-half partials via LDS; both sub-waves get full s
        #pragma unroll
        for (int r = 0; r < 8; ++r) xch[wave][r][lane] = acc[r];
        __syncthreads();
        #pragma unroll
        for (int r = 0; r < 8; ++r) acc[r] += xch[wave ^ 1][r][lane];
        __syncthreads();

        // ---------- squash over d (16 lanes of a half-wave) ----------
        #pragma unroll
        for (int r = 0; r < 8; ++r) {
            float s2    = half16_sum(acc[r] * acc[r]);
            float scale = s2 / ((1.0f + s2) * sqrtf(s2 + CAPS_EPS));
            vcap[r]     = scale * acc[r];
        }

        if (it < ROUTINGS - 1) {
            // ---------- agreement: blog[m,i,n] (+)= <u_hat[m,i,n,:], v[m,:]> ----------
            for (int i = i0; i < i1; ++i) {
                const float* ax = xp + i * DIN;
                v2f a0 = *(const v2f*)(ax);
                v2f a1 = *(const v2f*)(ax + 4);
                const float* aw = wp + (size_t)i * (NOUT * DOUT * DIN);
                v2f w0 = *(const v2f*)(aw);
                v2f w1 = *(const v2f*)(aw + 4);
                v8f u = {};
                u = __builtin_amdgcn_wmma_f32_16x16x4_f32(
                        false, a0, false, w0, (short)0, u, false, false);
                u = __builtin_amdgcn_wmma_f32_16x16x4_f32(
                        false, a1, false, w1, (short)0, u, false, false);
                float t[8];
                #pragma unroll
                for (int r = 0; r < 8; ++r) t[r] = u[r] * vcap[r];
                // 8-shuffle butterfly: lane ends with full d-sum for row ridx
                float val = butterfly8x16(t, lane);
                if (lm < 8) {
                    size_t bi = (size_t)(i * NOUT + n) * MT + (ridx + half * 8);
                    if (it == 0) blog[bi] = val;      // b starts at zero
                    else         blog[bi] += val;
                }
            }
        } else if (sub == 0) {
            // ---------- final v -> out[b0+m, n, d] ----------
            #pragma unroll
            for (int r = 0; r < 8; ++r) {
                int m = r + half * 8;
                out[((size_t)(b0 + m) * NOUT + n) * DOUT + lm] = vcap[r];
            }
        }
    }
}

extern "C" void kernel_launch(void* const* d_in, const int* in_sizes, int n_in,
                              void* d_out, int out_size, void* d_ws, size_t ws_size,
                              hipStream_t stream) {
    (void)in_sizes; (void)n_in; (void)out_size; (void)ws_size;
    const float* x = (const float*)d_in[0];     // [1024, 1152, 8]
    const float* W = (const float*)d_in[1];     // [1, 1152, 10, 16, 8]
    float* out  = (float*)d_out;                // [1024, 10, 16]
    float* blog = (float*)d_ws;                 // needs 64*1152*10*16*4 B = 47.2 MB
    dim3 grid(B_TOT / MT), block(NTHREADS);
    hipLaunchKernelGGL(caps_route, grid, block, 0, stream, x, W, out, blog);
}